// EarthMoverDistance_27865747817019
// MI455X (gfx1250) — compile-verified
//
#include <hip/hip_runtime.h>

// Fused approx-EMD for B=4, N=M=4096, 3-D points.
// dist tiles recomputed on the fly via V_WMMA_F32_16X16X4_F32:
//   A = [x, y, z, s1] (16x4), B = [-2x', -2y', -2z', 1] (4x16), C = s2 (bcast/col)
//   D = s1 + s2 - 2*cross  (16x16 f32 tile, 8 VGPRs/lane)
// Only O(B*N) state in d_ws; the 268MB dist/match matrices are never stored.

typedef float v2f __attribute__((ext_vector_type(2)));
typedef float v8f __attribute__((ext_vector_type(8)));

#define BATCH 4
#define NPTS 4096
#define TILE 16
#define NSTRIP (NPTS / TILE)                    // 256 strips of 16 rows/cols
#define WAVES_PER_BLOCK 4
#define BLOCK_THREADS (WAVES_PER_BLOCK * 32)    // 128 (wave32)
#define TOTAL_STRIPS (BATCH * NSTRIP)           // 1024
#define GRID_BLOCKS (TOTAL_STRIPS / WAVES_PER_BLOCK) // 256
#define EMD_EPS 1e-9f

// ---------------------------------------------------------------------------
// D = A*B + C with C = broadcast cval (s2 per column).
__device__ __forceinline__ v8f emd_dist_tile(v2f a, v2f b, float cval) {
    v8f c;
#pragma unroll
    for (int v = 0; v < 8; ++v) c[v] = cval;
    // 8-arg form: (neg_a, A, neg_b, B, c_mod, C, reuse_a, reuse_b)
    return __builtin_amdgcn_wmma_f32_16x16x4_f32(
        false, a, false, b, (short)0, c, false, false);
}

__device__ __forceinline__ void strip_decode(int& b, int& base, int& lane,
                                             int& half, int& sub) {
    int wave  = threadIdx.x >> 5;
    lane      = threadIdx.x & 31;
    half      = lane >> 4;     // 0: K=0,1 / rows 0..7 ; 1: K=2,3 / rows 8..15
    sub       = lane & 15;     // M (A) or N (B/C/D) index inside tile
    int strip = blockIdx.x * WAVES_PER_BLOCK + wave;
    b         = strip / NSTRIP;
    base      = (strip % NSTRIP) * TILE;
}

// ---------------------------------------------------------------------------
__global__ void __launch_bounds__(BLOCK_THREADS)
emd_init(float* remainL, float* remainR, float* ratioL, float* ratioR,
         float* costPartial) {
    int i = blockIdx.x * blockDim.x + threadIdx.x;
    if (i < BATCH * NPTS) {
        remainL[i] = 1.0f;   // multiL = 1 (n == m)
        remainR[i] = 1.0f;   // multiR = 1
        ratioL[i]  = 0.0f;
        ratioR[i]  = 0.0f;
    }
    if (i < TOTAL_STRIPS) costPartial[i] = 0.0f;
}

// Pass 1: per-row  suml_i = eps + sum_j e_ij * remainR_j ; ratioL = remainL/suml
__global__ void __launch_bounds__(BLOCK_THREADS)
emd_pass1(const float* __restrict__ xyz1, const float* __restrict__ xyz2,
          const float* __restrict__ remainL, const float* __restrict__ remainR,
          float* __restrict__ ratioL, float level) {
    int b, rbase, lane, half, sub;
    strip_decode(b, rbase, lane, half, sub);

    const float* x1 = xyz1 + (size_t)b * 3 * NPTS;
    const float* y1 = x1 + NPTS;
    const float* z1 = x1 + 2 * NPTS;
    const float* x2 = xyz2 + (size_t)b * 3 * NPTS;
    const float* y2 = x2 + NPTS;
    const float* z2 = x2 + 2 * NPTS;

    // A operand (rows fixed for this strip)
    int   r  = rbase + sub;
    float ax = x1[r], ay = y1[r], az = z1[r];
    float s1 = ax * ax + ay * ay + az * az;
    v2f a;
    a.x = half ? az : ax;
    a.y = half ? s1 : ay;

    float partial[8];
#pragma unroll
    for (int v = 0; v < 8; ++v) partial[v] = 0.0f;

    for (int jt = 0; jt < NSTRIP; ++jt) {
        int   cidx = jt * TILE + sub;
        float bx = x2[cidx], by = y2[cidx], bz = z2[cidx];
        float s2 = bx * bx + by * by + bz * bz;
        v2f bb;
        bb.x = half ? (-2.0f * bz) : (-2.0f * bx);
        bb.y = half ? 1.0f         : (-2.0f * by);

        v8f d = emd_dist_tile(a, bb, s2);
        float rR = remainR[b * NPTS + cidx];
#pragma unroll
        for (int v = 0; v < 8; ++v) {
            float dist = fmaxf(d[v], 0.0f);
            partial[v] += __expf(level * dist) * rR;
        }
    }
    // row-sum across the 16 lanes of each half (cols of the tile)
#pragma unroll
    for (int v = 0; v < 8; ++v) {
        float p = partial[v];
        p += __shfl_xor(p, 1, 16);
        p += __shfl_xor(p, 2, 16);
        p += __shfl_xor(p, 4, 16);
        p += __shfl_xor(p, 8, 16);
        partial[v] = p;
    }
    if (sub == 0) {
#pragma unroll
        for (int v = 0; v < 8; ++v) {
            int row = b * NPTS + rbase + half * 8 + v;
            ratioL[row] = remainL[row] / (EMD_EPS + partial[v]);
        }
    }
}

// Pass 2: per-col  sumr_j = remainR_j * sum_i e_ij * ratioL_i ;
//         ratioR_j = min(remainR/(sumr+eps),1)*remainR ; remainR' = max(0, remainR - sumr)
__global__ void __launch_bounds__(BLOCK_THREADS)
emd_pass2(const float* __restrict__ xyz1, const float* __restrict__ xyz2,
          const float* __restrict__ ratioL, float* __restrict__ remainR,
          float* __restrict__ ratioR, float level) {
    int b, cbase, lane, half, sub;
    strip_decode(b, cbase, lane, half, sub);

    const float* x1 = xyz1 + (size_t)b * 3 * NPTS;
    const float* y1 = x1 + NPTS;
    const float* z1 = x1 + 2 * NPTS;
    const float* x2 = xyz2 + (size_t)b * 3 * NPTS;
    const float* y2 = x2 + NPTS;
    const float* z2 = x2 + 2 * NPTS;

    // B operand + C (cols fixed for this strip)
    int   c  = cbase + sub;
    float bx = x2[c], by = y2[c], bz = z2[c];
    float s2 = bx * bx + by * by + bz * bz;
    v2f bb;
    bb.x = half ? (-2.0f * bz) : (-2.0f * bx);
    bb.y = half ? 1.0f         : (-2.0f * by);

    float cp = 0.0f;
    for (int it = 0; it < NSTRIP; ++it) {
        int   r  = it * TILE + sub;
        float ax = x1[r], ay = y1[r], az = z1[r];
        float s1 = ax * ax + ay * ay + az * az;
        v2f a;
        a.x = half ? az : ax;
        a.y = half ? s1 : ay;

        v8f d = emd_dist_tile(a, bb, s2);
#pragma unroll
        for (int v = 0; v < 8; ++v) {
            float dist = fmaxf(d[v], 0.0f);
            float rl   = ratioL[b * NPTS + it * TILE + half * 8 + v];
            cp += __expf(level * dist) * rl;
        }
    }
    cp += __shfl_xor(cp, 16, 32);   // merge row-halves; lanes 0-15 hold col sums
    if (half == 0) {
        int   col  = b * NPTS + cbase + sub;
        float rR   = remainR[col];
        float sumr = rR * cp;
        float cons = fminf(rR / (sumr + EMD_EPS), 1.0f);
        ratioR[col]  = cons * rR;
        remainR[col] = fmaxf(0.0f, rR - sumr);
    }
}

// Pass 3: w_ij = e_ij * ratioL_i * ratioR_j ;
//         cost += w * sqrt(dist) ; remainL_i = max(0, remainL_i - sum_j w)
__global__ void __launch_bounds__(BLOCK_THREADS)
emd_pass3(const float* __restrict__ xyz1, const float* __restrict__ xyz2,
          const float* __restrict__ ratioL, const float* __restrict__ ratioR,
          float* __restrict__ remainL, float* __restrict__ costPartial,
          float level) {
    int b, rbase, lane, half, sub;
    strip_decode(b, rbase, lane, half, sub);
    int strip = blockIdx.x * WAVES_PER_BLOCK + (threadIdx.x >> 5);

    const float* x1 = xyz1 + (size_t)b * 3 * NPTS;
    const float* y1 = x1 + NPTS;
    const float* z1 = x1 + 2 * NPTS;
    const float* x2 = xyz2 + (size_t)b * 3 * NPTS;
    const float* y2 = x2 + NPTS;
    const float* z2 = x2 + 2 * NPTS;

    int   r  = rbase + sub;
    float ax = x1[r], ay = y1[r], az = z1[r];
    float s1 = ax * ax + ay * ay + az * az;
    v2f a;
    a.x = half ? az : ax;
    a.y = half ? s1 : ay;

    float rl[8];
#pragma unroll
    for (int v = 0; v < 8; ++v)
        rl[v] = ratioL[b * NPTS + rbase + half * 8 + v];

    float rowsum[8];
#pragma unroll
    for (int v = 0; v < 8; ++v) rowsum[v] = 0.0f;
    float costp = 0.0f;

    for (int jt = 0; jt < NSTRIP; ++jt) {
        int   cidx = jt * TILE + sub;
        float bx = x2[cidx], by = y2[cidx], bz = z2[cidx];
        float s2 = bx * bx + by * by + bz * bz;
        v2f bb;
        bb.x = half ? (-2.0f * bz) : (-2.0f * bx);
        bb.y = half ? 1.0f         : (-2.0f * by);

        v8f d = emd_dist_tile(a, bb, s2);
        float rr = ratioR[b * NPTS + cidx];
#pragma unroll
        for (int v = 0; v < 8; ++v) {
            float dist = fmaxf(d[v], 0.0f);
            float w    = __expf(level * dist) * rl[v] * rr;
            rowsum[v] += w;
            costp += w * sqrtf(fmaxf(dist, 1e-20f));
        }
    }
#pragma unroll
    for (int v = 0; v < 8; ++v) {
        float p = rowsum[v];
        p += __shfl_xor(p, 1, 16);
        p += __shfl_xor(p, 2, 16);
        p += __shfl_xor(p, 4, 16);
        p += __shfl_xor(p, 8, 16);
        rowsum[v] = p;
    }
    if (sub == 0) {
#pragma unroll
        for (int v = 0; v < 8; ++v) {
            int row = b * NPTS + rbase + half * 8 + v;
            remainL[row] = fmaxf(0.0f, remainL[row] - rowsum[v]);
        }
    }
    costp += __shfl_xor(costp, 1, 32);
    costp += __shfl_xor(costp, 2, 32);
    costp += __shfl_xor(costp, 4, 32);
    costp += __shfl_xor(costp, 8, 32);
    costp += __shfl_xor(costp, 16, 32);
    if (lane == 0) costPartial[strip] += costp;   // one writer per strip
}

__global__ void __launch_bounds__(256)
emd_final(const float* __restrict__ costPartial, float* __restrict__ out) {
    __shared__ float sh[256];
    int t = threadIdx.x;
    float s = 0.0f;
    for (int i = t; i < TOTAL_STRIPS; i += 256) s += costPartial[i];
    sh[t] = s;
    __syncthreads();
    for (int off = 128; off > 0; off >>= 1) {
        if (t < off) sh[t] += sh[t + off];
        __syncthreads();
    }
    // mean over batches of cost_b / min(n,m)  ==  total / (NPTS * BATCH)
    if (t == 0) out[0] = sh[0] / ((float)NPTS * (float)BATCH);
}

extern "C" void kernel_launch(void* const* d_in, const int* in_sizes, int n_in,
                              void* d_out, int out_size, void* d_ws, size_t ws_size,
                              hipStream_t stream) {
    (void)in_sizes; (void)n_in; (void)out_size; (void)ws_size;
    const float* xyz1 = (const float*)d_in[0];   // (B,3,N) f32
    const float* xyz2 = (const float*)d_in[1];   // (B,3,N) f32
    float*       out  = (float*)d_out;           // scalar f32

    float* ws          = (float*)d_ws;
    float* remainL     = ws;                         // B*N
    float* remainR     = remainL + BATCH * NPTS;     // B*N
    float* ratioL      = remainR + BATCH * NPTS;     // B*N
    float* ratioR      = ratioL  + BATCH * NPTS;     // B*N
    float* costPartial = ratioR  + BATCH * NPTS;     // TOTAL_STRIPS

    emd_init<<<(BATCH * NPTS + BLOCK_THREADS - 1) / BLOCK_THREADS,
               BLOCK_THREADS, 0, stream>>>(remainL, remainR, ratioL, ratioR,
                                           costPartial);

    // level = -4^j for j = 7..-1, then 0  (10 sweeps)
    const float levels[10] = {-16384.0f, -4096.0f, -1024.0f, -256.0f, -64.0f,
                              -16.0f, -4.0f, -1.0f, -0.25f, 0.0f};
    for (int s = 0; s < 10; ++s) {
        float lv = levels[s];
        emd_pass1<<<GRID_BLOCKS, BLOCK_THREADS, 0, stream>>>(
            xyz1, xyz2, remainL, remainR, ratioL, lv);
        emd_pass2<<<GRID_BLOCKS, BLOCK_THREADS, 0, stream>>>(
            xyz1, xyz2, ratioL, remainR, ratioR, lv);
        emd_pass3<<<GRID_BLOCKS, BLOCK_THREADS, 0, stream>>>(
            xyz1, xyz2, ratioL, ratioR, remainL, costPartial, lv);
    }
    emd_final<<<1, 256, 0, stream>>>(costPartial, out);
}